// MeshConvTranspose_49263274885413
// MI455X (gfx1250) — compile-verified
//
#include <hip/hip_runtime.h>

typedef float v2f __attribute__((ext_vector_type(2)));
typedef float v8f __attribute__((ext_vector_type(8)));

#define NV_   40962
#define NF_   81920
#define NVC_  10242
#define B_    4
#define CIN_  32
#define COUT_ 32
#define KDIM_ 128               // CIN * NCOEFF
#define NCOL_ (B_ * NV_)        // 163848 GEMM columns
#define LSTR_ 132               // LDS row stride (pad: 132 % 64 == 4 -> no bank conflicts)

// ---------------------------------------------------------------------------
// Kernel 0: transpose x_coarse [b][cin][nvc] -> x_t [b][nvc][cin]
// ---------------------------------------------------------------------------
__global__ void k0_transpose(const float* __restrict__ x, float* __restrict__ xt) {
    int idx = blockIdx.x * 256 + threadIdx.x;       // coalesced read over v
    if (idx >= B_ * CIN_ * NVC_) return;
    int v = idx % NVC_;
    int t = idx / NVC_;
    int i = t % CIN_;
    int b = t / CIN_;
    xt[(b * NVC_ + v) * CIN_ + i] = x[idx];
}

// ---------------------------------------------------------------------------
// Kernel 1: per-face gradient fused with EW/NS dot.
//   gfe[b][f][c] = sum_d (sum_j Gv[(d*NF+f)*3+j] * x[b, c, Gc[...]]) * EW[f,d]
// Lanes = channel c: coalesced x gathers and gfe/gfn writes; G row data is
// wave-uniform (broadcast loads); batch loop reuses G/EW/NS registers 4x.
// ---------------------------------------------------------------------------
__global__ void k1_gradface(const float* __restrict__ xt,
                            const int*   __restrict__ Gc,
                            const float* __restrict__ Gv,
                            const float* __restrict__ NS,
                            const float* __restrict__ EW,
                            float* __restrict__ gfe,
                            float* __restrict__ gfn) {
    const int c = threadIdx.x;                              // 0..31
    const int f = blockIdx.x * blockDim.y + threadIdx.y;
    if (f >= NF_) return;

    int   col[9];
    float val[9];
#pragma unroll
    for (int d = 0; d < 3; ++d)
#pragma unroll
        for (int j = 0; j < 3; ++j) {
            int e = (d * NF_ + f) * 3 + j;
            col[d * 3 + j] = Gc[e];
            val[d * 3 + j] = Gv[e];
        }
    const float ew0 = EW[f * 3 + 0], ew1 = EW[f * 3 + 1], ew2 = EW[f * 3 + 2];
    const float ns0 = NS[f * 3 + 0], ns1 = NS[f * 3 + 1], ns2 = NS[f * 3 + 2];

#pragma unroll
    for (int b = 0; b < B_; ++b) {
        const float* xb = xt + (size_t)b * NVC_ * CIN_;
        float g[3];
#pragma unroll
        for (int d = 0; d < 3; ++d) {
            float s = 0.f;
#pragma unroll
            for (int j = 0; j < 3; ++j) {
                int cc = col[d * 3 + j];
                float xv = (cc < NVC_) ? xb[(size_t)cc * CIN_ + c] : 0.f;
                s = fmaf(val[d * 3 + j], xv, s);
            }
            g[d] = s;
        }
        size_t o = ((size_t)b * NF_ + f) * CIN_ + c;
        gfe[o] = fmaf(g[0], ew0, fmaf(g[1], ew1, g[2] * ew2));
        gfn[o] = fmaf(g[0], ns0, fmaf(g[1], ns1, g[2] * ns2));
    }
}

// ---------------------------------------------------------------------------
// Kernel 2 (fused feat + WMMA GEMM):
// Block = 128 threads (4 waves), 64 GEMM columns. Phase 1: each wave builds
// the 128-long feat vectors for its own 16 columns (lanes = channel i;
// coalesced gathers; one ds_store_b128 per lane per column into a
// 132-float-stride LDS row). Phase 2: the wave consumes its LDS tile as the
// WMMA B operand (ds_load_b64, conflict-free thanks to the padded stride)
// against coeffs[32x128] (L2-resident) -> out[32 x NCOL] + bias.
// EXEC stays all-ones through the WMMAs: tail columns are clamped for
// compute and only the final stores are guarded.
// ---------------------------------------------------------------------------
__global__ void k2_feat_wmma(const float* __restrict__ xt,
                             const int*   __restrict__ Lc,
                             const float* __restrict__ Lv,
                             const int*   __restrict__ Fc,
                             const float* __restrict__ Fv,
                             const float* __restrict__ gfe,
                             const float* __restrict__ gfn,
                             const float* __restrict__ coef,
                             const float* __restrict__ bias,
                             float* __restrict__ out) {
    __shared__ float lds[64 * LSTR_];

    const int lane    = threadIdx.x & 31;
    const int wave    = threadIdx.x >> 5;
    const int colbase = blockIdx.x * 64;

    // ---------------- Phase 1: build feat tiles in LDS -------------------
    const int i = lane;                                     // channel
#pragma unroll 1
    for (int t = 0; t < 16; ++t) {
        const int cl = wave * 16 + t;                       // local column
        int n = colbase + cl;
        if (n >= NCOL_) n = NCOL_ - 1;                      // clamp tail
        const int b = n / NV_;
        const int v = n % NV_;

        const float* xb = xt  + (size_t)b * NVC_ * CIN_;
        const float* ge = gfe + (size_t)b * NF_  * CIN_;
        const float* gn = gfn + (size_t)b * NF_  * CIN_;

        float xf = (v < NVC_) ? xb[(size_t)v * CIN_ + i] : 0.f;

        float lap = 0.f;
#pragma unroll
        for (int j = 0; j < 7; ++j) {
            int   cc = Lc[v * 7 + j];                       // wave-uniform
            float lv = Lv[v * 7 + j];
            float xv = (cc < NVC_) ? xb[(size_t)cc * CIN_ + i] : 0.f;
            lap = fmaf(lv, xv, lap);
        }
        float gv_ew = 0.f, gv_ns = 0.f;
#pragma unroll
        for (int j = 0; j < 6; ++j) {
            int   fc = Fc[v * 6 + j];                       // wave-uniform
            float fv = Fv[v * 6 + j];
            size_t o = (size_t)fc * CIN_ + i;
            gv_ew = fmaf(fv, ge[o], gv_ew);
            gv_ns = fmaf(fv, gn[o], gv_ns);
        }
        float4 fv4 = make_float4(xf, lap, gv_ew, gv_ns);    // K = i*4 + {0..3}
        *(float4*)(lds + cl * LSTR_ + i * 4) = fv4;         // ds_store_b128
    }
    __syncthreads();

    // ---------------- Phase 2: WMMA GEMM from LDS ------------------------
    const int koff = (lane >> 4) * 2;                       // 0 or 2
    const int m    = lane & 15;

    const float* lp  = lds  + (wave * 16 + (lane & 15)) * LSTR_ + koff;
    const float* a0p = coef + (size_t)m        * KDIM_ + koff;
    const float* a1p = coef + (size_t)(m + 16) * KDIM_ + koff;

    v8f c0 = {};
    v8f c1 = {};
#pragma unroll 4
    for (int kb = 0; kb < KDIM_; kb += 4) {
        v2f bv = *(const v2f*)(lp  + kb);                   // ds_load_b64
        v2f a0 = *(const v2f*)(a0p + kb);
        v2f a1 = *(const v2f*)(a1p + kb);
        c0 = __builtin_amdgcn_wmma_f32_16x16x4_f32(false, a0, false, bv,
                                                   (short)0, c0, false, false);
        c1 = __builtin_amdgcn_wmma_f32_16x16x4_f32(false, a1, false, bv,
                                                   (short)0, c1, false, false);
    }

    const int ncol = colbase + wave * 16 + (lane & 15);
    if (ncol < NCOL_) {
        const int b    = ncol / NV_;
        const int v    = ncol % NV_;
        const int half = (lane >> 4) * 8;                   // D rows: M = r + half
#pragma unroll
        for (int r = 0; r < 8; ++r) {
            int o0 = r + half;                              // rows 0..15
            out[((size_t)(b * COUT_ + o0)) * NV_ + v] = c0[r] + bias[o0];
            int o1 = 16 + r + half;                         // rows 16..31
            out[((size_t)(b * COUT_ + o1)) * NV_ + v] = c1[r] + bias[o1];
        }
    }
}

// ---------------------------------------------------------------------------
extern "C" void kernel_launch(void* const* d_in, const int* in_sizes, int n_in,
                              void* d_out, int out_size, void* d_ws, size_t ws_size,
                              hipStream_t stream) {
    (void)in_sizes; (void)n_in; (void)out_size; (void)ws_size;

    const float* x_coarse = (const float*)d_in[0];
    const float* coeffs   = (const float*)d_in[1];
    const float* bias     = (const float*)d_in[2];
    const float* NS       = (const float*)d_in[3];
    const float* EW       = (const float*)d_in[4];
    const int*   G_cols   = (const int*)  d_in[6];
    const float* G_vals   = (const float*)d_in[7];
    const int*   L_cols   = (const int*)  d_in[9];
    const float* L_vals   = (const float*)d_in[10];
    const int*   F_cols   = (const int*)  d_in[12];
    const float* F_vals   = (const float*)d_in[13];
    float* out = (float*)d_out;

    // workspace layout (floats): ~89 MB total, L2-resident working set
    float* xt  = (float*)d_ws;                              // B*NVC*CIN
    float* gfe = xt  + (size_t)B_ * NVC_ * CIN_;            // B*NF*CIN
    float* gfn = gfe + (size_t)B_ * NF_  * CIN_;            // B*NF*CIN

    // k0: transpose x to channel-last
    {
        int total = B_ * CIN_ * NVC_;
        k0_transpose<<<(total + 255) / 256, 256, 0, stream>>>(x_coarse, xt);
    }
    // k1: per-face gradient + EW/NS projection
    {
        dim3 blk(32, 8);
        dim3 grd((NF_ + 7) / 8);
        k1_gradface<<<grd, blk, 0, stream>>>(xt, G_cols, G_vals, NS, EW, gfe, gfn);
    }
    // k2: fused feat construction (LDS) + WMMA GEMM + bias
    {
        dim3 blk(128);                                      // 4 waves / block
        dim3 grd((NCOL_ + 63) / 64);
        k2_feat_wmma<<<grd, blk, 0, stream>>>(xt, L_cols, L_vals, F_cols, F_vals,
                                              gfe, gfn, coeffs, bias, out);
    }
}